// InputEmbedder_64450279244460
// MI455X (gfx1250) — compile-verified
//
#include <hip/hip_runtime.h>

// ---------------------------------------------------------------------------
// InputEmbedder for MI455X (gfx1250).
//   out[i][j][e] = a[i][e] + b[j][e] + ptab[clip(ri[i]-ri[j],-32,32)+32][e]
// Output is 1024*1024*128 fp32 = 512 MiB -> purely store-bandwidth bound
// (~23 us at 23.3 TB/s).  Phase-1 GEMMs (~11 MFLOP) done with
// v_wmma_f32_16x16x4_f32; phase-2 is a float4 NT-store stream with 4 rows
// per wave to amortize address math and the a[i] load.
// ---------------------------------------------------------------------------

typedef __attribute__((ext_vector_type(2))) float v2f;
typedef __attribute__((ext_vector_type(4))) float v4f;
typedef __attribute__((ext_vector_type(8))) float v8f;

#define N_SEQ   1024
#define IN_DIM  21
#define E_DIM   128
#define BINS    32
#define NBUCKET 65          // 2*BINS + 1

// ---------------------------------------------------------------------------
// Phase 1a: a = tf @ Wa^T + ba ; b = tf @ Wb^T + bb  via WMMA f32 16x16x4.
// One wave per 16x16 output tile.  K=21 padded to 24 (6 steps of 4).
// blockIdx.y selects which matrix (0 -> a, 1 -> b).
// ---------------------------------------------------------------------------
__global__ void proj_wmma_kernel(const float* __restrict__ tf,
                                 const float* __restrict__ Wa,
                                 const float* __restrict__ ba,
                                 const float* __restrict__ Wb,
                                 const float* __restrict__ bb,
                                 float* __restrict__ out_a,
                                 float* __restrict__ out_b)
{
    const int lane = threadIdx.x & 31;
    const int wave = threadIdx.x >> 5;
    const int lin  = blockIdx.x * 8 + wave;          // tile id, 0..511
    const int tm   = (lin >> 3) * 16;                // output row base (M)
    const int tn   = (lin & 7) * 16;                 // output col base (N/E)

    const float* W    = blockIdx.y ? Wb : Wa;        // [E_DIM][IN_DIM]
    const float* bias = blockIdx.y ? bb : ba;        // [E_DIM]
    float*       out  = blockIdx.y ? out_b : out_a;  // [N_SEQ][E_DIM]

    const int half = lane >> 4;                      // 0: K={0,1}, 1: K={2,3}
    const int l16  = lane & 15;
    const int arow = tm + l16;                       // A row this lane feeds
    const int bcol = tn + l16;                       // B column this lane feeds

    v8f acc = {};
    #pragma unroll
    for (int kk = 0; kk < 6; ++kk) {
        const int k0 = kk * 4 + half * 2;
        v2f av, bv;
        av.x = (k0     < IN_DIM) ? tf[arow * IN_DIM + k0]     : 0.0f;
        av.y = (k0 + 1 < IN_DIM) ? tf[arow * IN_DIM + k0 + 1] : 0.0f;
        bv.x = (k0     < IN_DIM) ? W [bcol * IN_DIM + k0]     : 0.0f;
        bv.y = (k0 + 1 < IN_DIM) ? W [bcol * IN_DIM + k0 + 1] : 0.0f;
        // D = A(16x4) x B(4x16) + C   -> v_wmma_f32_16x16x4_f32
        acc = __builtin_amdgcn_wmma_f32_16x16x4_f32(
                  /*neg_a=*/false, av, /*neg_b=*/false, bv,
                  /*c_mod=*/(short)0, acc, /*reuse_a=*/false, /*reuse_b=*/false);
    }

    // C/D layout: VGPR r holds M = tm + r + 8*half, N = tn + (lane&15).
    const float bval  = bias[bcol];
    const int   mbase = tm + half * 8;
    #pragma unroll
    for (int r = 0; r < 8; ++r) {
        out[(mbase + r) * E_DIM + bcol] = acc[r] + bval;
    }
}

// ---------------------------------------------------------------------------
// Phase 1b: ptab[k][e] = Wp[e][k] + bp[e]   (65 x 128 fp32 = 33 KB)
// ---------------------------------------------------------------------------
__global__ void ptab_kernel(const float* __restrict__ Wp,   // [E_DIM][NBUCKET]
                            const float* __restrict__ bp,   // [E_DIM]
                            float* __restrict__ ptab)       // [NBUCKET][E_DIM]
{
    const int k = blockIdx.x;        // 0..64
    const int e = threadIdx.x;       // 0..127
    ptab[k * E_DIM + e] = Wp[e * NBUCKET + k] + bp[e];
}

// ---------------------------------------------------------------------------
// Phase 2: the 512 MiB broadcast stream.
// One wave per 4 consecutive (i, j) rows; lane handles a float4 of E.
// a[i] float4 is loaded once per wave and reused for all 4 rows; the 4 NT
// b128 stores land in one contiguous 2 KB span (clause-friendly immediate
// offsets).  NT keeps the write-once stream from rinsing L2; the small
// a/b/ptab reads stay RT and L2/L0-resident.
// ---------------------------------------------------------------------------
__global__ void broadcast_kernel(const float* __restrict__ a,
                                 const float* __restrict__ b,
                                 const float* __restrict__ ptab,
                                 const int*   __restrict__ ri,
                                 float* __restrict__ out)
{
    const int lane  = threadIdx.x & 31;
    const int jw    = threadIdx.x >> 5;              // 0..7
    const int i     = blockIdx.y;
    const int jbase = blockIdx.x * 32 + jw * 4;      // 4 rows per wave

    const int rii = ri[i];
    const v4f av  = ((const v4f*)(a + (size_t)i * E_DIM))[lane];

    float* obase = out + ((size_t)i * N_SEQ + (size_t)jbase) * E_DIM;

    #pragma unroll
    for (int jj = 0; jj < 4; ++jj) {
        const int j = jbase + jj;

        int delta = rii - ri[j];
        delta = (delta < -BINS) ? -BINS : ((delta > BINS) ? BINS : delta);
        delta += BINS;                               // 0..64

        const v4f bv = ((const v4f*)(b    + (size_t)j     * E_DIM))[lane];
        const v4f pv = ((const v4f*)(ptab + (size_t)delta * E_DIM))[lane];

        v4f r = av + bv + pv;
        __builtin_nontemporal_store(r, (v4f*)(obase + jj * E_DIM) + lane);
    }
}

// ---------------------------------------------------------------------------
extern "C" void kernel_launch(void* const* d_in, const int* in_sizes, int n_in,
                              void* d_out, int out_size, void* d_ws, size_t ws_size,
                              hipStream_t stream)
{
    const float* tf = (const float*)d_in[0];   // [1024][21]
    const int*   ri = (const int*)  d_in[1];   // [1024]
    const float* Wa = (const float*)d_in[2];   // [128][21]
    const float* ba = (const float*)d_in[3];   // [128]
    const float* Wb = (const float*)d_in[4];   // [128][21]
    const float* bb = (const float*)d_in[5];   // [128]
    const float* Wp = (const float*)d_in[6];   // [128][65]
    const float* bp = (const float*)d_in[7];   // [128]
    float* out = (float*)d_out;                // [1024][1024][128]

    // Workspace: a (512 KB) | b (512 KB) | ptab (33 KB)
    float* a    = (float*)d_ws;
    float* b    = a + (size_t)N_SEQ * E_DIM;
    float* ptab = b + (size_t)N_SEQ * E_DIM;

    // Phase 1: projections via WMMA (512 tiles per matrix, 8 waves/block).
    proj_wmma_kernel<<<dim3(64, 2), 256, 0, stream>>>(tf, Wa, ba, Wb, bb, a, b);

    // Phase 1b: fused position table (Wp^T + bp).
    ptab_kernel<<<dim3(NBUCKET), E_DIM, 0, stream>>>(Wp, bp, ptab);

    // Phase 2: 512 MiB NT-store stream (the entire runtime, ~23 us roofline).
    // 32768 blocks x 8 waves x 4 rows = 1024 x 1024 rows.
    broadcast_kernel<<<dim3(N_SEQ / 32, N_SEQ), 256, 0, stream>>>(a, b, ptab, ri, out);
}